// GCNModel_16776142258488
// MI455X (gfx1250) — compile-verified
//
#include <hip/hip_runtime.h>

typedef __attribute__((ext_vector_type(2))) float v2f;
typedef __attribute__((ext_vector_type(8))) float v8f;

// ---------------------------------------------------------------------------
// WMMA GEMM: C[M,128] = A[M,128] @ W[128,128] (+ optional bias broadcast)
// Requires M % 16 == 0.  block = 128 threads (4 waves), each wave owns one
// 16-wide N tile; grid = (M/16, 2)  ->  8 N tiles of 16 cover N = 128.
// Uses V_WMMA_F32_16X16X4_F32 (full fp32 precision), 32 MMAs per tile.
// ---------------------------------------------------------------------------
__global__ void __launch_bounds__(128)
wmma_gemm_128(const float* __restrict__ A, const float* __restrict__ W,
              const float* __restrict__ bias, float* __restrict__ C, int M) {
  const int wave  = threadIdx.x >> 5;
  const int lane  = threadIdx.x & 31;
  const int half  = lane >> 4;   // 0: lanes 0-15, 1: lanes 16-31
  const int l16   = lane & 15;
  const int mtile = blockIdx.x;
  const int ntile = blockIdx.y * 4 + wave;

  const float* Arow = A + (size_t)(mtile * 16 + l16) * 128;
  const float* Wcol = W + ntile * 16 + l16;

  v8f acc = {};
#pragma unroll
  for (int kb = 0; kb < 128; kb += 4) {
    const int k0 = kb + 2 * half;
    // A fragment (16x4 f32): lane holds A[m=l16][k0], A[m=l16][k0+1]
    v2f a = *(const v2f*)(Arow + k0);
    // B fragment (4x16 f32): lane holds W[k0][n=l16], W[k0+1][n=l16]
    v2f b;
    b.x = Wcol[(size_t)(k0 + 0) * 128];
    b.y = Wcol[(size_t)(k0 + 1) * 128];
    // 8 args: (neg_a, A, neg_b, B, c_mod, C, reuse_a, reuse_b)
    acc = __builtin_amdgcn_wmma_f32_16x16x4_f32(false, a, false, b,
                                                (short)0, acc, false, false);
  }

  const float bv = bias ? bias[ntile * 16 + l16] : 0.0f;
  float* Crow = C + (size_t)(mtile * 16) * 128 + ntile * 16 + l16;
#pragma unroll
  for (int r = 0; r < 8; ++r)
    Crow[(size_t)(r + 8 * half) * 128] = acc[r] + bv;
}

// ---------------------------------------------------------------------------
// Degree / normalization kernels
// ---------------------------------------------------------------------------
__global__ void init_deg(float* __restrict__ deg, int n) {
  int i = blockIdx.x * blockDim.x + threadIdx.x;
  if (i < n) deg[i] = 1.0f;  // self-loop contributes 1 to in-degree
}

__global__ void count_deg(const long long* __restrict__ dst, int E,
                          float* __restrict__ deg) {
  int e = blockIdx.x * blockDim.x + threadIdx.x;
  if (e < E) atomicAdd(&deg[(int)dst[e]], 1.0f);
}

__global__ void finish_dinv(float* __restrict__ deg, int n) {
  int i = blockIdx.x * blockDim.x + threadIdx.x;
  if (i < n) {
    float d = deg[i];
    deg[i] = (d > 0.0f) ? rsqrtf(d) : 0.0f;
  }
}

// ---------------------------------------------------------------------------
// agg[i][c] = hW[i][c] * dinv[i]^2   (self-loop term; also zero-initializes)
// ---------------------------------------------------------------------------
__global__ void selfloop_init(const float* __restrict__ hW,
                              const float* __restrict__ dinv,
                              float* __restrict__ agg, int total) {
  int i = blockIdx.x * blockDim.x + threadIdx.x;
  if (i < total) {
    float di = dinv[i >> 7];  // node index = i / 128
    agg[i] = hW[i] * (di * di);
  }
}

// ---------------------------------------------------------------------------
// Edge scatter-add: agg[dst] += hW[src] * dinv[src]*dinv[dst]
// 32 threads per edge, each thread handles 4 channels (float4 gather,
// 4x global_atomic_add_f32 scatter).  Working set is L2-resident (25.6 MB).
// ---------------------------------------------------------------------------
__global__ void edge_agg(const float* __restrict__ hW,
                         const float* __restrict__ dinv,
                         const long long* __restrict__ src,
                         const long long* __restrict__ dst,
                         int E, float* __restrict__ agg) {
  long long gid = (long long)blockIdx.x * blockDim.x + threadIdx.x;
  int e = (int)(gid >> 5);
  int c4 = ((int)gid & 31) * 4;
  if (e >= E) return;
  int s = (int)src[e];
  int d = (int)dst[e];
  float norm = dinv[s] * dinv[d];
  float4 v = *(const float4*)(hW + (size_t)s * 128 + c4);
  float* op = agg + (size_t)d * 128 + c4;
  atomicAdd(op + 0, v.x * norm);
  atomicAdd(op + 1, v.y * norm);
  atomicAdd(op + 2, v.z * norm);
  atomicAdd(op + 3, v.w * norm);
}

// ---------------------------------------------------------------------------
// out[i] = (agg[i] + bias[i % 128])  with optional ReLU
// ---------------------------------------------------------------------------
__global__ void bias_act(const float* __restrict__ in,
                         const float* __restrict__ bias,
                         float* __restrict__ out, int total, int do_relu) {
  int i = blockIdx.x * blockDim.x + threadIdx.x;
  if (i < total) {
    float v = in[i] + bias[i & 127];
    out[i] = do_relu ? fmaxf(v, 0.0f) : v;
  }
}

// ---------------------------------------------------------------------------
extern "C" void kernel_launch(void* const* d_in, const int* in_sizes, int n_in,
                              void* d_out, int out_size, void* d_ws, size_t ws_size,
                              hipStream_t stream) {
  const float*     x  = (const float*)d_in[0];
  const long long* ei = (const long long*)d_in[1];   // int64 [2, E]
  const float*     q  = (const float*)d_in[2];
  const float*     W1 = (const float*)d_in[3];
  const float*     b1 = (const float*)d_in[4];
  const float*     W2 = (const float*)d_in[5];
  const float*     b2 = (const float*)d_in[6];
  const float*     Wq = (const float*)d_in[7];
  const float*     bq = (const float*)d_in[8];

  const int N = in_sizes[0] / 128;   // 50000 nodes
  const int E = in_sizes[1] / 2;     // 800000 edges
  const int Q = in_sizes[2] / 128;   // 20000 queries

  const long long* src = ei;
  const long long* dst = ei + E;

  float* out  = (float*)d_out;
  float* ques = out;                     // [Q,128] first in tuple order
  float* h2   = out + (size_t)Q * 128;   // [N,128] second

  // Workspace: dinv [N] + bufA [N*128].  The h2 slot of d_out doubles as
  // buffer B (fully overwritten with the final h2 before returning).
  float* wsf  = (float*)d_ws;
  float* dinv = wsf;
  float* bufA = wsf + (((size_t)N + 255) & ~(size_t)255);
  float* bufB = h2;

  const int T = 256;
  const int total = N * 128;
  const long long ethreads = (long long)E * 32;
  const int egrid = (int)((ethreads + T - 1) / T);
  dim3 blk128(128);
  dim3 gN(N / 16, 2);
  dim3 gQ(Q / 16, 2);

  // --- deg^{-1/2} --------------------------------------------------------
  init_deg   <<<(N + T - 1) / T, T, 0, stream>>>(dinv, N);
  count_deg  <<<(E + T - 1) / T, T, 0, stream>>>(dst, E, dinv);
  finish_dinv<<<(N + T - 1) / T, T, 0, stream>>>(dinv, N);

  // --- layer 1: h1 = relu(aggregate(x @ W1) + b1) ------------------------
  wmma_gemm_128<<<gN, blk128, 0, stream>>>(x, W1, nullptr, bufA, N);
  selfloop_init<<<(total + T - 1) / T, T, 0, stream>>>(bufA, dinv, bufB, total);
  edge_agg     <<<egrid, T, 0, stream>>>(bufA, dinv, src, dst, E, bufB);
  bias_act     <<<(total + T - 1) / T, T, 0, stream>>>(bufB, b1, bufA, total, 1);

  // --- layer 2: h2 = aggregate(h1 @ W2) + b2 -----------------------------
  wmma_gemm_128<<<gN, blk128, 0, stream>>>(bufA, W2, nullptr, bufB, N);
  selfloop_init<<<(total + T - 1) / T, T, 0, stream>>>(bufB, dinv, bufA, total);
  edge_agg     <<<egrid, T, 0, stream>>>(bufB, dinv, src, dst, E, bufA);
  bias_act     <<<(total + T - 1) / T, T, 0, stream>>>(bufA, b2, h2, total, 0);

  // --- query projection: ques = q_emb @ Wq + bq --------------------------
  wmma_gemm_128<<<gQ, blk128, 0, stream>>>(q, Wq, bq, ques, Q);
}